// GraphConvolution_70677981823578
// MI455X (gfx1250) — compile-verified
//
#include <hip/hip_runtime.h>

typedef __attribute__((ext_vector_type(2))) float v2f;
typedef __attribute__((ext_vector_type(4))) float v4f;
typedef __attribute__((ext_vector_type(8))) float v8f;

#define KCH 2
#define ZDIM 128
#define FDIM 128

// ---------------------------------------------------------------------------
// Kernel 0: zero the accumulation buffer (d_out) with 16B stores.
// ---------------------------------------------------------------------------
__global__ void gcn_zero_kernel(float* __restrict__ out, int n4) {
    int i = blockIdx.x * blockDim.x + threadIdx.x;
    if (i < n4) {
        v4f z = {0.0f, 0.0f, 0.0f, 0.0f};
        *(v4f*)(out + (size_t)i * 4) = z;
    }
}

// ---------------------------------------------------------------------------
// Kernel 1: support[k] = x[k] @ W using V_WMMA_F32_16X16X4_F32.
// Block = 256 threads (8 waves), covering 128 nodes of one K-channel.
//
// W is staged in LDS *pair-packed*: wlds[p*128 + f] = (W[2p][f], W[2p+1][f])
// so each lane's B fragment (two K-adjacent elements of one F column) is a
// single aligned ds_load_b64 into an even VGPR pair — no v_mov regrouping,
// and all 8 fragments of a z-step issue before a single s_wait_dscnt.
//
// WMMA f32 16x16x4 lane layout (ISA 7.12.2):
//   A 16x4 : lanes 0-15 M=lane, K0/K1 in VGPR0/1; lanes 16-31 K2/K3.
//   B 4x16 : lanes 0-15 N=lane, K0/K1;            lanes 16-31 K2/K3.
//   C 16x16: VGPR r = row r (lanes 0-15) / row r+8 (lanes 16-31), N = lane&15.
// ---------------------------------------------------------------------------
__global__ void gcn_gemm_wmma_kernel(const float* __restrict__ x,
                                     const float* __restrict__ w,
                                     float* __restrict__ support,
                                     int N) {
    __shared__ v2f wlds[(ZDIM / 2) * FDIM];        // 64 KB, pair-packed

    const int tid = threadIdx.x;
    // Cooperative staging: coalesced 16B global reads, pair-packed LDS writes.
    {
        float* wsc = (float*)wlds;                 // scalar view of LDS
        for (int i = tid * 4; i < ZDIM * FDIM; i += 256 * 4) {
            const int z  = i >> 7;                 // W row
            const int f  = i & 127;                // W col
            const int p  = z >> 1;                 // K-pair index
            const int lo = z & 1;                  // slot within the pair
            v4f r4 = *(const v4f*)(w + i);
            wsc[(((p * FDIM) + f + 0) << 1) | lo] = r4.x;
            wsc[(((p * FDIM) + f + 1) << 1) | lo] = r4.y;
            wsc[(((p * FDIM) + f + 2) << 1) | lo] = r4.z;
            wsc[(((p * FDIM) + f + 3) << 1) | lo] = r4.w;
        }
    }
    __syncthreads();

    const int k    = blockIdx.y;
    const int lane = tid & 31;
    const int wv   = tid >> 5;
    const int n0   = blockIdx.x * 128 + wv * 16;   // first node row of this wave
    const int m    = lane & 15;                    // row/col within 16
    const int pb   = lane >> 4;                    // 0: K0/K1 half, 1: K2/K3 half

    int nA = n0 + m;
    if (nA >= N) nA = N - 1;                       // clamp (stores are guarded)
    const float* __restrict__ xrow = x + ((size_t)k * N + nA) * ZDIM;

    v8f acc[8] = {};                               // 8 tiles of 16x16 F columns

    #pragma unroll
    for (int z0 = 0; z0 < ZDIM; z0 += 4) {
        v2f a = *(const v2f*)(xrow + z0 + (pb << 1));          // A frag (8B)
        const v2f* __restrict__ brow = wlds + ((z0 >> 1) + pb) * FDIM + m;
        #pragma unroll
        for (int t = 0; t < 8; ++t) {
            v2f b = brow[t * 16];                              // one ds_load_b64
            acc[t] = __builtin_amdgcn_wmma_f32_16x16x4_f32(
                false, a, false, b, (short)0, acc[t], false, false);
        }
    }

    const int rowOff = pb * 8;
    #pragma unroll
    for (int t = 0; t < 8; ++t) {
        #pragma unroll
        for (int r = 0; r < 8; ++r) {
            int row = n0 + r + rowOff;
            if (row < N) {
                support[((size_t)k * N + row) * FDIM + t * 16 + m] = acc[t][r];
            }
        }
    }
}

// ---------------------------------------------------------------------------
// Kernel 2: COO SpMM scatter. One wave per (k, edge):
//   out[k][row][f] += val * support[k][col][f]   for f = 0..127
// 4 coalesced 128B gathers (support row lives in L2: 51MB/channel << 192MB L2)
// and 4 coalesced bursts of global_atomic_add_f32.
// ---------------------------------------------------------------------------
__global__ void gcn_spmm_kernel(const int* __restrict__ rows,
                                const int* __restrict__ cols,
                                const float* __restrict__ vals,
                                const float* __restrict__ support,
                                float* __restrict__ out,
                                int E, int N) {
    const int gid  = blockIdx.x * blockDim.x + threadIdx.x;
    const int e    = gid >> 5;
    const int lane = gid & 31;
    if (e >= E) return;
    const int k = blockIdx.y;

    const int   r = rows[e];
    const int   c = cols[e];
    const float v = vals[e];

    const float* __restrict__ src = support + ((size_t)k * N + c) * FDIM;
    float* __restrict__       dst = out     + ((size_t)k * N + r) * FDIM;

    #pragma unroll
    for (int i = 0; i < 4; ++i) {
        const int f = i * 32 + lane;               // coalesced 128B per burst
        unsafeAtomicAdd(dst + f, v * src[f]);      // -> global_atomic_add_f32
    }
}

// ---------------------------------------------------------------------------
// Kernel 3: in-place ReLU on d_out.
// ---------------------------------------------------------------------------
__global__ void gcn_relu_kernel(float* __restrict__ out, int n4) {
    int i = blockIdx.x * blockDim.x + threadIdx.x;
    if (i < n4) {
        v4f v = *(v4f*)(out + (size_t)i * 4);
        v.x = fmaxf(v.x, 0.0f);
        v.y = fmaxf(v.y, 0.0f);
        v.z = fmaxf(v.z, 0.0f);
        v.w = fmaxf(v.w, 0.0f);
        *(v4f*)(out + (size_t)i * 4) = v;
    }
}

// ---------------------------------------------------------------------------
// Launch: zero d_out -> WMMA GEMM into d_ws -> atomic scatter -> ReLU.
// ---------------------------------------------------------------------------
extern "C" void kernel_launch(void* const* d_in, const int* in_sizes, int n_in,
                              void* d_out, int out_size, void* d_ws, size_t ws_size,
                              hipStream_t stream) {
    const float* x        = (const float*)d_in[0];   // [K, N, Z]
    const int*   adj_rows = (const int*)  d_in[1];   // [E]
    const int*   adj_cols = (const int*)  d_in[2];   // [E]
    const float* adj_vals = (const float*)d_in[3];   // [E]
    const float* weight   = (const float*)d_in[4];   // [Z, F]

    const int E = in_sizes[1];
    const int N = in_sizes[0] / (KCH * ZDIM);        // 100000

    float* out     = (float*)d_out;
    float* support = (float*)d_ws;                   // K*N*F f32 = 102.4 MB

    // 0) zero the accumulator
    {
        int n4 = out_size / 4;                       // out_size = K*N*F, /4 exact
        int blocks = (n4 + 255) / 256;
        gcn_zero_kernel<<<blocks, 256, 0, stream>>>(out, n4);
    }
    // 1) dense GEMM: support = x @ W (per channel)
    {
        dim3 grid((N + 127) / 128, KCH);
        gcn_gemm_wmma_kernel<<<grid, 256, 0, stream>>>(x, weight, support, N);
    }
    // 2) sparse scatter-add: one wave per (k, edge)
    {
        int blocks = (int)(((long long)E * 32 + 255) / 256);
        dim3 grid(blocks, KCH);
        gcn_spmm_kernel<<<grid, 256, 0, stream>>>(adj_rows, adj_cols, adj_vals,
                                                  support, out, E, N);
    }
    // 3) ReLU in place
    {
        int n4 = out_size / 4;
        int blocks = (n4 + 255) / 256;
        gcn_relu_kernel<<<blocks, 256, 0, stream>>>(out, n4);
    }
}